// TokenTransform3D_75402445849017
// MI455X (gfx1250) — compile-verified
//
#include <hip/hip_runtime.h>
#include <hip/hip_bf16.h>

// VQ nearest-neighbour: x = weights - condition (flat; token n = x[32n:32n+32]),
// indices[n] = argmin_k ||x_n - e_k||^2 = argmax_k (x_n . e_k - 0.5*|e_k|^2)
// GEMM [N x 32] @ [32 x 1024]^T via V_WMMA_F32_16X16X4_F32 (full fp32), codebook
// staged in LDS (stride-34 padded rows -> 64-bank conflict-free ds_load_b64).
// Each wave owns TWO 16-token M tiles sharing each B fragment: two independent
// WMMA accumulator chains interleave (hides XDL latency) and halve LDS traffic.

typedef __attribute__((ext_vector_type(2))) float v2f;
typedef __attribute__((ext_vector_type(8))) float v8f;

#define DIMK       32
#define NCODES     1024
#define LDS_STRIDE 34          // 34 floats/row: 8B aligned, 64-bank conflict-free
#define BLOCK      256         // 8 waves
#define TOK_PER_WAVE 32        // two 16-row WMMA M tiles per wave

__launch_bounds__(BLOCK, 2)
__global__ void vq_argmin_wmma_kernel(const float* __restrict__ w,
                                      const float* __restrict__ cond,
                                      const float* __restrict__ cb,
                                      int* __restrict__ out, int ntok) {
  __shared__ float cb_lds[NCODES * LDS_STRIDE];  // 136 KB
  __shared__ float hnorm[NCODES];                //   4 KB

  const int tid = threadIdx.x;

  // ---- stage codebook into LDS (coalesced float4 global loads) ----
  for (int i = tid; i < NCODES * (DIMK / 4); i += BLOCK) {
    const int row = i >> 3;        // 8 float4 per 32-float row
    const int c4  = i & 7;
    const float4 v = ((const float4*)cb)[i];
    float* dst = &cb_lds[row * LDS_STRIDE + c4 * 4];
    dst[0] = v.x; dst[1] = v.y; dst[2] = v.z; dst[3] = v.w;
  }
  __syncthreads();
  for (int e = tid; e < NCODES; e += BLOCK) {
    const float* r = &cb_lds[e * LDS_STRIDE];
    float s = 0.f;
#pragma unroll
    for (int k = 0; k < DIMK; ++k) s += r[k] * r[k];
    hnorm[e] = 0.5f * s;
  }
  __syncthreads();

  // ---- per-wave: two 16-token M tiles ----
  const int wave   = tid >> 5;                 // wave32
  const int lane   = tid & 31;
  const int mrow   = lane & 15;                // M row (A) / N column (B,C,D)
  const int half   = lane >> 4;                // lane group 0/1
  const int m_base = (blockIdx.x * (BLOCK / 32) + wave) * TOK_PER_WAVE;
  if (m_base >= ntok) return;                  // wave-uniform

  // A fragments (16x4 f32): lane l holds X[m][4k+2*(l>>4)], [+1]; m = l&15
  v2f a0[8], a1[8];
  {
    const size_t r0 = (size_t)(m_base + mrow) * DIMK;
    const size_t r1 = (size_t)(m_base + 16 + mrow) * DIMK;
#pragma unroll
    for (int k = 0; k < 8; ++k) {
      const int j = 4 * k + 2 * half;
      a0[k][0] = w[r0 + j]     - cond[r0 + j];
      a0[k][1] = w[r0 + j + 1] - cond[r0 + j + 1];
      a1[k][0] = w[r1 + j]     - cond[r1 + j];
      a1[k][1] = w[r1 + j + 1] - cond[r1 + j + 1];
    }
  }

  float bv0[8], bv1[8];
  int   bi0[8], bi1[8];
#pragma unroll
  for (int r = 0; r < 8; ++r) {
    bv0[r] = -__builtin_inff(); bi0[r] = 0;
    bv1[r] = -__builtin_inff(); bi1[r] = 0;
  }

  // ---- sweep all 1024 codes in 16-wide N tiles ----
  for (int base = 0; base < NCODES; base += 16) {
    const int col = base + mrow;               // this lane's code column
    const float nh = -hnorm[col];
    v8f c0 = {nh, nh, nh, nh, nh, nh, nh, nh}; // C init = -0.5*|e_col|^2
    v8f c1 = c0;

    const float* brow = &cb_lds[col * LDS_STRIDE + 2 * half];
#pragma unroll
    for (int k = 0; k < 8; ++k) {
      // B fragment 4x16: lane l holds E[col][4k+2*(l>>4)], [+1]  (ds_load_b64)
      const v2f b = *(const v2f*)(brow + 4 * k);
      c0 = __builtin_amdgcn_wmma_f32_16x16x4_f32(false, a0[k], false, b,
                                                 (short)0, c0, false, false);
      c1 = __builtin_amdgcn_wmma_f32_16x16x4_f32(false, a1[k], false, b,
                                                 (short)0, c1, false, false);
    }

    // running argmax; C/D layout: vgpr r -> M=r (lanes 0-15) / M=r+8 (16-31)
#pragma unroll
    for (int r = 0; r < 8; ++r) {
      const float v0 = c0[r];
      if (v0 > bv0[r]) { bv0[r] = v0; bi0[r] = col; }
      const float v1 = c1[r];
      if (v1 > bv1[r]) { bv1[r] = v1; bi1[r] = col; }
    }
  }

  // ---- reduce across the 16 lanes of each half-wave (columns of each row) ----
#pragma unroll
  for (int off = 1; off < 16; off <<= 1) {
#pragma unroll
    for (int r = 0; r < 8; ++r) {
      float ov = __shfl_xor(bv0[r], off, 32);
      int   oi = __shfl_xor(bi0[r], off, 32);
      if (ov > bv0[r] || (ov == bv0[r] && oi < bi0[r])) { bv0[r] = ov; bi0[r] = oi; }
      ov = __shfl_xor(bv1[r], off, 32);
      oi = __shfl_xor(bi1[r], off, 32);
      if (ov > bv1[r] || (ov == bv1[r] && oi < bi1[r])) { bv1[r] = ov; bi1[r] = oi; }
    }
  }
  if (mrow == 0) {
    const int m0 = m_base + 8 * half;          // lanes 0 and 16 write 8 rows each
#pragma unroll
    for (int r = 0; r < 8; ++r) {
      out[m0 + r]      = bi0[r];
      out[m0 + 16 + r] = bi1[r];
    }
  }
}

extern "C" void kernel_launch(void* const* d_in, const int* in_sizes, int n_in,
                              void* d_out, int out_size, void* d_ws, size_t ws_size,
                              hipStream_t stream) {
  (void)n_in; (void)d_ws; (void)ws_size; (void)out_size;
  const float* weights   = (const float*)d_in[0];
  const float* condition = (const float*)d_in[1];
  const float* codebook  = (const float*)d_in[2];
  int* out = (int*)d_out;                       // reference output: int32 indices

  const int ntok    = in_sizes[0] / DIMK;       // 4194304 / 32 = 131072
  const int tok_per_block = (BLOCK / 32) * TOK_PER_WAVE;  // 256
  const int nblocks = (ntok + tok_per_block - 1) / tok_per_block;

  vq_argmin_wmma_kernel<<<nblocks, BLOCK, 0, stream>>>(weights, condition,
                                                       codebook, out, ntok);
}